// TotalVariationMinimization_37563783971538
// MI455X (gfx1250) — compile-verified
//
#include <hip/hip_runtime.h>

// TV minimization, [64,3,512,512] f32, 10 iterations fused via temporal
// blocking in LDS. Forward-difference stencil => halo needed only on the
// +x/+y sides: a 64x64 LDS tile yields a valid 54x54 output after 10 iters.
// Tile load uses the CDNA5 Tensor Data Mover (tensor_load_to_lds, TENSORcnt),
// with descriptor padding producing the 68-float LDS row stride and OOB
// zero-fill replacing software address clamps.

#define IMG_H 512
#define IMG_W 512
#define OUT_T 54          // valid output per tile edge
#define TILE  64          // working tile edge (OUT_T + 10 halo)
#define LSTR  68          // LDS row stride in floats (272B; TDM pad 64+4 DW)
#define N_ITER 10
#define TV_W  0.1f
#define TV_EPS 1e-8f

typedef unsigned int u32;
typedef u32 u32x4 __attribute__((ext_vector_type(4)));
typedef u32 u32x8 __attribute__((ext_vector_type(8)));

__device__ __forceinline__ unsigned lds_byte_off(const void* p) {
  // generic pointer to LDS: low 32 bits are the LDS byte address
  return (unsigned)(unsigned long long)p;
}

__device__ __forceinline__ void wait_tensor0() {
#if __has_builtin(__builtin_amdgcn_s_wait_tensorcnt)
  __builtin_amdgcn_s_wait_tensorcnt(0);
#else
  asm volatile("s_wait_tensorcnt 0x0" ::: "memory");
#endif
}

// Issue one TDM 2D tile load: global (tile origin) -> LDS, 64x64 f32,
// tensor row stride 512, LDS row padding 64+4 DWORDs, OOB reads -> 0.
__device__ __forceinline__ void tdm_tile_load(const float* g_tile_origin,
                                              unsigned lds_addr,
                                              u32 rem_x, u32 rem_y) {
  const unsigned long long ga = (unsigned long long)g_tile_origin;

  u32x4 g0;
  g0.x = 0x1u;                                   // count=1 (valid), user desc
  g0.y = lds_addr;                               // lds_addr [63:32]
  g0.z = (u32)ga;                                // global_addr [95:64]
  g0.w = (u32)((ga >> 32) & 0x01FFFFFFu)         // global_addr [120:96]
         | (2u << 30);                           // type=2 ("image")

  u32x8 g1;
  g1.s0 = (0u)                                   // workgroup_mask=0 (no cluster)
        | (2u << 16)                             // data_size: 4 bytes
        | (1u << 20)                             // pad_enable
        | (5u << 22)                             // pad_interval: 64 DWORDs
        | (3u << 25);                            // pad_amount: 4 DWORDs
  g1.s1 = (rem_x & 0xFFFFu) << 16;               // abar_addr=0 | tensor_dim0 lo
  g1.s2 = (rem_x >> 16) | ((rem_y & 0xFFFFu) << 16); // dim0 hi | tensor_dim1 lo
  g1.s3 = (rem_y >> 16) | ((u32)TILE << 16);     // dim1 hi | tile_dim0=64
  g1.s4 = (u32)TILE;                             // tile_dim1=64, tile_dim2=0
  g1.s5 = (u32)IMG_W;                            // tensor_dim0_stride lo32 = 512
  g1.s6 = 0u;                                    // stride hi | dim1_stride lo
  g1.s7 = 0u;                                    // dim1_stride hi (unused, 2D)

  // VADDR2/VADDR3 omitted (NULL) => tensor up to 2D
  asm volatile("tensor_load_to_lds %0, %1" :: "s"(g0), "s"(g1) : "memory");
}

__global__ __launch_bounds__(256)
void tv_min_fused_kernel(const float* __restrict__ x, float* __restrict__ out) {
  __shared__ float s_orig[TILE][LSTR];
  __shared__ float s_a[TILE][LSTR];
  __shared__ float s_b[TILE][LSTR];

  const int tx  = threadIdx.x;            // 0..63
  const int ty  = threadIdx.y;            // 0..3
  const int tid = ty * 64 + tx;           // 0..255
  const int gx0 = blockIdx.x * OUT_T;
  const int gy0 = blockIdx.y * OUT_T;
  const long long plane = (long long)blockIdx.z * (IMG_H * IMG_W);
  const float* xp = x + plane;

  // ---- TDM tile load: one DMA per workgroup (wave 0 only; TDM ignores EXEC,
  // so other waves must branch around it). OOB rows/cols zero-fill, which
  // only touches halo slack outside the valid 54x54 output cone.
  if (tid == 0) {
    tdm_tile_load(xp + (gy0 * IMG_W + gx0),
                  lds_byte_off(&s_orig[0][0]),
                  (u32)(IMG_W - gx0), (u32)(IMG_H - gy0));
    wait_tensor0();
  }
  __syncthreads();

  float (*cur)[LSTR] = s_orig;   // iter 0 reads original directly
  float (*nxt)[LSTR] = s_a;

  for (int it = 0; it < N_ITER; ++it) {
#pragma unroll
    for (int r = ty; r < TILE; r += 4) {
      const int gi = gy0 + r;
      const int gj = gx0 + tx;
      const float c = cur[r][tx];
      const int jp = (tx < TILE - 1) ? tx + 1 : tx;   // clamp inside tile
      const int rp = (r  < TILE - 1) ? r  + 1 : r;
      // reference: dx/dy are forward diffs, zero-padded at trailing edge
      const float right = (gj + 1 < IMG_W) ? cur[r][jp] : c;
      const float down  = (gi + 1 < IMG_H) ? cur[rp][tx] : c;
      const float dx = right - c;
      const float dy = down  - c;
      const float g  = (dx + dy) * rsqrtf(dx * dx + dy * dy + TV_EPS);
      float t = c - TV_W * g;                 // gradient step
      t = 0.9f * t + 0.1f * s_orig[r][tx];    // blend toward original
      t = fminf(fmaxf(t, 0.0f), 1.0f);        // clip
      nxt[r][tx] = t;
    }
    __syncthreads();
    cur = nxt;
    nxt = (cur == s_a) ? s_b : s_a;           // s_orig preserved for blending
  }

  // ---- store valid 54x54 region (tiles stride by 54 => exact cover, no overlap)
  if (tx < OUT_T) {
    const int gj = gx0 + tx;
    if (gj < IMG_W) {
#pragma unroll
      for (int r = ty; r < OUT_T; r += 4) {
        const int gi = gy0 + r;
        if (gi < IMG_H) out[plane + gi * IMG_W + gj] = cur[r][tx];
      }
    }
  }
}

extern "C" void kernel_launch(void* const* d_in, const int* in_sizes, int n_in,
                              void* d_out, int out_size, void* d_ws, size_t ws_size,
                              hipStream_t stream) {
  const float* x = (const float*)d_in[0];
  float* out = (float*)d_out;
  const int planes = in_sizes[0] / (IMG_H * IMG_W);   // 64*3 = 192
  dim3 grid((IMG_W + OUT_T - 1) / OUT_T,              // 10
            (IMG_H + OUT_T - 1) / OUT_T,              // 10
            planes);                                  // 192
  dim3 block(64, 4, 1);                               // 8 wave32
  hipLaunchKernelGGL(tv_min_fused_kernel, grid, block, 0, stream, x, out);
}